// HeteroMLPPredictor_558345749135
// MI455X (gfx1250) — compile-verified
//
#include <hip/hip_runtime.h>

typedef __attribute__((ext_vector_type(2))) float v2f;
typedef __attribute__((ext_vector_type(4))) float v4f;
typedef __attribute__((ext_vector_type(8))) float v8f;

#define DIMS 128
#define CHN  16

// Kernel 1: per-node projection via f32 WMMA.
//   P[n][c]      = sum_k h[n][k] * W[c][k]        (c = 0..15,  W_s part)
//   P[n][16 + c] = sum_k h[n][k] * W[c][128 + k]  (c = 0..15,  W_d part)
// One wave -> one 16x16 output tile, K=128 via 32x V_WMMA_F32_16X16X4_F32.
__global__ __launch_bounds__(256) void node_proj_wmma(
    const float* __restrict__ h, const float* __restrict__ W,
    float* __restrict__ P, int n_nodes, int n_tiles_m)
{
    const int wave = blockIdx.x * (blockDim.x >> 5) + (threadIdx.x >> 5);
    const int lane = threadIdx.x & 31;
    const int total_tiles = n_tiles_m * 2;
    if (wave >= total_tiles) return;   // whole-wave uniform exit: EXEC all-1s for WMMA

    const int m_tile    = wave >> 1;
    const int n_tile    = wave & 1;
    const int node_base = m_tile * 16;
    const int row       = lane & 15;   // M (for A) / N (for B)
    const int half      = lane >> 4;   // K-pair select: 0 -> K{0,1}, 1 -> K{2,3}

    // A fragment source: h row (clamped for generality at the tail)
    int node = node_base + row;
    if (node >= n_nodes) node = n_nodes - 1;
    const float* aptr = h + (size_t)node * DIMS + half * 2;

    // B fragment source: B[k][n] = Wcat[n][k]; Wcat rows alias W directly.
    const int c = n_tile * 16 + row;
    const float* bptr =
        W + (c < CHN ? (size_t)c * (2 * DIMS)
                     : (size_t)(c - CHN) * (2 * DIMS) + DIMS) + half * 2;

    v8f acc = {};
#pragma unroll
    for (int kk = 0; kk < DIMS / 4; ++kk) {
        v2f a  = *(const v2f*)(aptr + kk * 4);   // {K, K+1} for this lane's half
        v2f bb = *(const v2f*)(bptr + kk * 4);
        acc = __builtin_amdgcn_wmma_f32_16x16x4_f32(
            /*neg_a=*/false, a, /*neg_b=*/false, bb,
            /*c_mod=*/(short)0, acc, /*reuse_a=*/false, /*reuse_b=*/false);
    }

    // D layout (32-bit C/D 16x16): VGPR r, lanes 0-15 -> M=r, lanes 16-31 -> M=r+8
#pragma unroll
    for (int r = 0; r < 8; ++r) {
        const int out_row = node_base + half * 8 + r;
        if (out_row < n_nodes)
            P[(size_t)out_row * 32 + n_tile * 16 + row] = acc[r];
    }
}

// Kernel 2: out[e][c] = P[src[e]][c] + P[dst[e]][16+c] + b[c]
// 4 threads per edge, float4 per thread. P (12.8 MB) stays L2-resident;
// output is streamed with nontemporal stores so it doesn't evict P.
__global__ __launch_bounds__(256) void edge_combine(
    const long long* __restrict__ src, const long long* __restrict__ dst,
    const float* __restrict__ P, const float* __restrict__ b,
    float* __restrict__ out, int n_edges)
{
    const int gid = blockIdx.x * blockDim.x + threadIdx.x;
    const int e = gid >> 2;
    if (e >= n_edges) return;
    const int q = gid & 3;

    const long long s  = src[e];
    const long long d0 = dst[e];

    v4f ps = *(const v4f*)(P + (size_t)s  * 32 +      q * 4);
    v4f pd = *(const v4f*)(P + (size_t)d0 * 32 + 16 + q * 4);
    v4f vb = *(const v4f*)(b + q * 4);

    v4f o = ps + pd + vb;
    __builtin_nontemporal_store(o, (v4f*)(out + (size_t)e * CHN + q * 4));
}

extern "C" void kernel_launch(void* const* d_in, const int* in_sizes, int n_in,
                              void* d_out, int out_size, void* d_ws, size_t ws_size,
                              hipStream_t stream) {
    const float*     h   = (const float*)d_in[0];
    const long long* src = (const long long*)d_in[1];   // int64 per reference
    const long long* dst = (const long long*)d_in[2];
    const float*     W   = (const float*)d_in[3];       // (16, 256): [W_s | W_d]
    const float*     b   = (const float*)d_in[4];
    float* out = (float*)d_out;
    float* P   = (float*)d_ws;                          // needs n_nodes*32*4 = 12.8 MB

    const int n_nodes = in_sizes[0] / DIMS;
    const int n_edges = in_sizes[1];

    // Phase 1: node projections (WMMA GEMM). 2 waves of tiles per 16 nodes.
    const int n_tiles_m = (n_nodes + 15) / 16;
    const int waves     = n_tiles_m * 2;
    const int blocks1   = (waves + 7) / 8;              // 8 waves per 256-thread block
    node_proj_wmma<<<blocks1, 256, 0, stream>>>(h, W, P, n_nodes, n_tiles_m);

    // Phase 2: edge gather + add (HBM-bound, ~130 MB total).
    const long long threads2 = (long long)n_edges * 4;
    const int blocks2 = (int)((threads2 + 255) / 256);
    edge_combine<<<blocks2, 256, 0, stream>>>(src, dst, P, b, out, n_edges);
}